// MHA_KNN_15960098472025
// MI455X (gfx1250) — compile-verified
//
#include <hip/hip_runtime.h>

typedef __attribute__((ext_vector_type(2))) float v2f;
typedef __attribute__((ext_vector_type(8))) float v8f;

constexpr int B_  = 8;
constexpr int V_  = 4096;
constexpr int E_  = 256;
constexpr int K_  = 6;
constexpr int NP_ = B_ * V_;          // 32768 points

// ---------------- WMMA helper: D = A(16x4 f32) * B(4x16 f32) + C ----------------
__device__ __forceinline__ v8f wmma4(v2f a, v2f b, v8f c) {
  // 8 args: (neg_a, A, neg_b, B, c_mod, C, reuse_a, reuse_b)
  return __builtin_amdgcn_wmma_f32_16x16x4_f32(false, a, false, b, (short)0, c,
                                               false, false);
}

// ---------------- squared norms: one wave per point ----------------
__global__ __launch_bounds__(256) void sqnorm_kernel(const float* __restrict__ x,
                                                     float* __restrict__ sq) {
  const int lane = threadIdx.x & 31;
  const int p = blockIdx.x * (blockDim.x >> 5) + (threadIdx.x >> 5);
  if (p >= NP_) return;
  const float* row = x + (size_t)p * E_ + lane * 8;
  float4 a = *(const float4*)row;
  float4 b = *(const float4*)(row + 4);
  float s = a.x*a.x + a.y*a.y + a.z*a.z + a.w*a.w
          + b.x*b.x + b.y*b.y + b.z*b.z + b.w*b.w;
#pragma unroll
  for (int m = 16; m >= 1; m >>= 1) s += __shfl_xor(s, m, 32);
  if (lane == 0) sq[p] = s;
}

// ---------------- GEMM: C[M,E] = A[M,E] * W[E,E]^T (+ residual) ----------------
// one wave per 16x64 tile: A fragment loaded once per k-step, reused 4x
template <bool RES>
__global__ __launch_bounds__(256) void gemm_nt_wmma(const float* __restrict__ A,
                                                    const float* __restrict__ W,
                                                    const float* __restrict__ residual,
                                                    float* __restrict__ C, int M) {
  const int lane   = threadIdx.x & 31;
  const int wave   = blockIdx.x * (blockDim.x >> 5) + (threadIdx.x >> 5);
  const int tilesN = E_ / 64;                       // 4 column-tiles of 64
  const int mBase  = (wave / tilesN) * 16;
  const int nBase  = (wave % tilesN) * 64;
  if (mBase >= M) return;
  const int lrow = lane & 15, lhi = lane >> 4;
  const float* aP = A + (size_t)(mBase + lrow) * E_ + 2 * lhi;   // A frag
  const float* wP = W + (size_t)(nBase + lrow) * E_ + 2 * lhi;   // B = W^T frag
  v8f acc[4] = {{}, {}, {}, {}};
#pragma unroll 4
  for (int k0 = 0; k0 < E_; k0 += 4) {
    v2f av = *(const v2f*)(aP + k0);
#pragma unroll
    for (int t = 0; t < 4; ++t) {
      v2f bv = *(const v2f*)(wP + k0 + t * 16 * E_);
      acc[t] = wmma4(av, bv, acc[t]);
    }
  }
#pragma unroll
  for (int t = 0; t < 4; ++t) {
#pragma unroll
    for (int r = 0; r < 8; ++r) {
      const int m = mBase + r + 8 * lhi;           // C layout: VGPR r, lane-half lhi
      const size_t off = (size_t)m * E_ + nBase + 16 * t + lrow;
      float vv = acc[t][r];
      if (RES) vv += residual[off];
      C[off] = vv;
    }
  }
}

// ---------------- KNN: WMMA Gram tiles + streaming per-row top-6 ----------------
__device__ __forceinline__ void topk_insert(float v, int j, float tv[K_], int ti[K_]) {
  if (v < tv[K_ - 1]) {
    tv[K_ - 1] = v; ti[K_ - 1] = j;
#pragma unroll
    for (int s = K_ - 1; s > 0; --s) {
      if (tv[s] < tv[s - 1]) {
        float fv = tv[s]; tv[s] = tv[s - 1]; tv[s - 1] = fv;
        int   fi = ti[s]; ti[s] = ti[s - 1]; ti[s - 1] = fi;
      }
    }
  }
}

// each wave: 16 rows x 64-col Gram super-tile per iteration (4 accumulators)
__global__ __launch_bounds__(256) void knn_kernel(const float* __restrict__ x,
                                                  const float* __restrict__ sq,
                                                  int* __restrict__ idxOut) {
  __shared__ float tileS[8][16][65];        // 16x64 d2-key tile per wave (padded)
  __shared__ float mvS[8][16][2][K_];       // per-half sorted top-6 values
  __shared__ int   miS[8][16][2][K_];       // per-half sorted top-6 indices

  const int waveId = threadIdx.x >> 5;
  const int lane   = threadIdx.x & 31;
  const int b      = blockIdx.x >> 5;                       // V_/128 = 32 blocks/batch
  const int rowBlk = (blockIdx.x & 31) * 128 + waveId * 16; // 16 rows per wave
  const float* xb  = x + (size_t)b * V_ * E_;

  const int lrow = lane & 15, lhi = lane >> 4;
  const float* aP = xb + (size_t)(rowBlk + lrow) * E_ + 2 * lhi;

  float tv[K_]; int ti[K_];
#pragma unroll
  for (int t = 0; t < K_; ++t) { tv[t] = 3.0e38f; ti[t] = 0; }

  const int c0 = lhi * 32;   // this lane scans column-half [c0, c0+32) of row lrow

  for (int jt = 0; jt < V_; jt += 64) {
    const float* bP = xb + (size_t)(jt + lrow) * E_ + 2 * lhi;
    v8f acc[4] = {{}, {}, {}, {}};
#pragma unroll 4
    for (int k0 = 0; k0 < E_; k0 += 4) {
      v2f av = *(const v2f*)(aP + k0);
#pragma unroll
      for (int t = 0; t < 4; ++t) {
        v2f bv = *(const v2f*)(bP + k0 + t * 16 * E_);
        acc[t] = wmma4(av, bv, acc[t]);
      }
    }
    // key(i,j) = sq[j] - 2*G(i,j)   (sq[i] is row-constant: ordering unchanged)
#pragma unroll
    for (int t = 0; t < 4; ++t) {
      const float sqj = sq[(size_t)b * V_ + jt + 16 * t + lrow];
#pragma unroll
      for (int r = 0; r < 8; ++r)
        tileS[waveId][r + 8 * lhi][16 * t + lrow] = sqj - 2.0f * acc[t][r];
    }
    // same-wave DS ops are in order: scan my row-half
#pragma unroll 8
    for (int c = 0; c < 32; ++c) {
      float vv = tileS[waveId][lrow][c0 + c];
      topk_insert(vv, jt + c0 + c, tv, ti);
    }
  }

  // merge the two column-half top-6 lists per row
#pragma unroll
  for (int t = 0; t < K_; ++t) {
    mvS[waveId][lrow][lhi][t] = tv[t];
    miS[waveId][lrow][lhi][t] = ti[t];
  }
  if (lane < 16) {
    int ia = 0, ib = 0;
    int* outp = idxOut + ((size_t)b * V_ + rowBlk + lane) * K_;
#pragma unroll
    for (int t = 0; t < K_; ++t) {
      float va = mvS[waveId][lane][0][ia];
      float vb = mvS[waveId][lane][1][ib];
      if (va <= vb) { outp[t] = miS[waveId][lane][0][ia]; ++ia; }
      else          { outp[t] = miS[waveId][lane][1][ib]; ++ib; }
    }
  }
}

// ---------------- zero the scatter-add target (re-run safe for graph replay) ----
__global__ void zero_kernel(float* __restrict__ p, int n) {
  int i = blockIdx.x * blockDim.x + threadIdx.x;
  if (i < n) p[i] = 0.0f;
}

// ---------------- per-point attention: one wave per point ----------------
// lane l holds elements [8l, 8l+8) of the E=256 row -> head = l/4 (HD=32)
__global__ __launch_bounds__(256) void attn_kernel(const float* __restrict__ Q,
                                                   const float* __restrict__ Kp,
                                                   const float* __restrict__ Vp,
                                                   const int* __restrict__ idx6,
                                                   float* __restrict__ Hout,
                                                   float* __restrict__ xw) {
  const int lane = threadIdx.x & 31;
  const int p = blockIdx.x * (blockDim.x >> 5) + (threadIdx.x >> 5);
  if (p >= NP_) return;
  const int b = p / V_;
  const float scale = 0.1767766952966369f;  // 1/sqrt(32)

  const float* qr = Q + (size_t)p * E_ + lane * 8;
  float q[8];
  { float4 a = *(const float4*)qr, c = *(const float4*)(qr + 4);
    q[0]=a.x; q[1]=a.y; q[2]=a.z; q[3]=a.w; q[4]=c.x; q[5]=c.y; q[6]=c.z; q[7]=c.w; }
  const float* vcr = Vp + (size_t)p * E_ + lane * 8;
  float vc[8];
  { float4 a = *(const float4*)vcr, c = *(const float4*)(vcr + 4);
    vc[0]=a.x; vc[1]=a.y; vc[2]=a.z; vc[3]=a.w; vc[4]=c.x; vc[5]=c.y; vc[6]=c.z; vc[7]=c.w; }

  int nbr[K_];
  const int* ip = idx6 + (size_t)p * K_;
#pragma unroll
  for (int k = 0; k < K_; ++k) nbr[k] = ip[k];

  float s[K_];
#pragma unroll
  for (int k = 0; k < K_; ++k) {
    const float* kr = Kp + (size_t)(b * V_ + nbr[k]) * E_ + lane * 8;
    float4 a = *(const float4*)kr, c = *(const float4*)(kr + 4);
    float d = q[0]*a.x + q[1]*a.y + q[2]*a.z + q[3]*a.w
            + q[4]*c.x + q[5]*c.y + q[6]*c.z + q[7]*c.w;
    d += __shfl_xor(d, 1, 32);       // reduce across the 4 lanes of this head
    d += __shfl_xor(d, 2, 32);
    s[k] = d * scale;
  }
  float mx = s[0];
#pragma unroll
  for (int k = 1; k < K_; ++k) mx = fmaxf(mx, s[k]);
  float a6[K_], sum = 0.0f;
#pragma unroll
  for (int k = 0; k < K_; ++k) { a6[k] = __expf(s[k] - mx); sum += a6[k]; }
  const float inv = 1.0f / sum;
#pragma unroll
  for (int k = 0; k < K_; ++k) a6[k] *= inv;

  float acc[8] = {0,0,0,0,0,0,0,0};
#pragma unroll
  for (int k = 0; k < K_; ++k) {
    const float* vr = Vp + (size_t)(b * V_ + nbr[k]) * E_ + lane * 8;
    float4 a = *(const float4*)vr, c = *(const float4*)(vr + 4);
    const float w = a6[k];
    acc[0] += w * (a.x - vc[0]); acc[1] += w * (a.y - vc[1]);
    acc[2] += w * (a.z - vc[2]); acc[3] += w * (a.w - vc[3]);
    acc[4] += w * (c.x - vc[4]); acc[5] += w * (c.y - vc[5]);
    acc[6] += w * (c.z - vc[6]); acc[7] += w * (c.w - vc[7]);
  }
  float* hr = Hout + (size_t)p * E_ + lane * 8;   // Q row fully consumed: alias-safe
  *(float4*)hr       = make_float4(acc[0], acc[1], acc[2], acc[3]);
  *(float4*)(hr + 4) = make_float4(acc[4], acc[5], acc[6], acc[7]);

  // x_w scatter: mean over heads = (full-wave sum of 4x-duplicated head weights)/32
#pragma unroll
  for (int k = 0; k < K_; ++k) {
    float w = a6[k];
#pragma unroll
    for (int m = 16; m >= 1; m >>= 1) w += __shfl_xor(w, m, 32);
    if (lane == 0) atomicAdd(xw + (size_t)b * V_ + nbr[k], w * (1.0f / 32.0f));
  }
}

// ---------------- host-side launcher ----------------
extern "C" void kernel_launch(void* const* d_in, const int* in_sizes, int n_in,
                              void* d_out, int out_size, void* d_ws, size_t ws_size,
                              hipStream_t stream) {
  const float* x  = (const float*)d_in[0];
  const float* Wq = (const float*)d_in[1];
  const float* Wk = (const float*)d_in[2];
  const float* Wv = (const float*)d_in[3];
  const float* Wo = (const float*)d_in[4];

  float* out = (float*)d_out;                       // [NP, E] residual output
  float* xw  = out + (size_t)NP_ * E_;              // [NP] scatter-added weights

  float* Qb  = (float*)d_ws;                        // also reused as H (attn out)
  float* Kb  = Qb + (size_t)NP_ * E_;
  float* Vb  = Kb + (size_t)NP_ * E_;
  float* sqb = Vb + (size_t)NP_ * E_;
  int*   idb = (int*)(sqb + NP_);

  const int waves = (NP_ / 16) * (E_ / 64);         // 8192 16x64 tiles per GEMM
  const int gemmBlocks = waves / 8;                 // 8 waves per block

  sqnorm_kernel<<<NP_ / 8, 256, 0, stream>>>(x, sqb);
  gemm_nt_wmma<false><<<gemmBlocks, 256, 0, stream>>>(x, Wq, nullptr, Qb, NP_);
  gemm_nt_wmma<false><<<gemmBlocks, 256, 0, stream>>>(x, Wk, nullptr, Kb, NP_);
  gemm_nt_wmma<false><<<gemmBlocks, 256, 0, stream>>>(x, Wv, nullptr, Vb, NP_);
  knn_kernel<<<B_ * (V_ / 128), 256, 0, stream>>>(x, sqb, idb);
  zero_kernel<<<(NP_ + 255) / 256, 256, 0, stream>>>(xw, NP_);
  attn_kernel<<<NP_ / 8, 256, 0, stream>>>(Qb, Kb, Vb, idb, Qb, xw);
  gemm_nt_wmma<true><<<gemmBlocks, 256, 0, stream>>>(Qb, Wo, x, out, NP_);
}